// QFNetBlock_86157043957988
// MI455X (gfx1250) — compile-verified
//
#include <hip/hip_runtime.h>
#include <stdint.h>

// MI455X (gfx1250) — memory-bound complex GEMM pipeline using fp32 WMMA +
// CDNA5 async global->LDS copies (ASYNCcnt) for the streaming U matrices.
// 512 MB of U matrices @ 23.3 TB/s dominates; each U element is read once.

typedef float v2f __attribute__((ext_vector_type(2)));
typedef float v8f __attribute__((ext_vector_type(8)));

#define DDIM 4096   // state dimension (S*E)
#define NBAT 256    // batch
#define BK   16     // K-chunk staged through LDS
#define LDU  20     // padded LDS row stride (floats) for U tiles  [16][LDU]
#define LDSS 20     // padded LDS k stride  (floats) for S tiles   [256][LDSS]
#define NW   12     // wires (2^12 = 4096)

// Async copy: note ISA adds the same IOFFSET to BOTH the global and LDS
// address.
#define ASYNC_CP128(ldsoff, voff, sbase, IMM)                              \
  asm volatile("global_load_async_to_lds_b128 %0, %1, %2 offset:" #IMM    \
               ::"v"(ldsoff), "v"(voff), "s"(sbase) : "memory")
#define ASYNC_CP64(ldsoff, voff, sbase, IMM)                               \
  asm volatile("global_load_async_to_lds_b64 %0, %1, %2 offset:" #IMM     \
               ::"v"(ldsoff), "v"(voff), "s"(sbase) : "memory")
#define WAIT_ASYNC0() asm volatile("s_wait_asynccnt 0" ::: "memory")

static __device__ __forceinline__ v8f wmma_f32_4(v2f a, v2f b, v8f c) {
  // V_WMMA_F32_16X16X4_F32 : D = A(16x4) * B(4x16) + C(16x16), fp32 throughout
  return __builtin_amdgcn_wmma_f32_16x16x4_f32(false, a, false, b, (short)0, c,
                                               false, false);
}

// ---------------------------------------------------------------- normalize
__global__ __launch_bounds__(256) void norm_kernel(const float* __restrict__ x,
                                                   float* __restrict__ s0) {
  __shared__ float red[256];
  const int b = blockIdx.x;
  const int t = threadIdx.x;
  const float* row = x + (size_t)b * DDIM;
  float v[16];
  float acc = 0.f;
#pragma unroll
  for (int q = 0; q < 16; ++q) {
    v[q] = row[t + (q << 8)];
    acc += v[q] * v[q];
  }
  red[t] = acc;
  __syncthreads();
  for (int off = 128; off > 0; off >>= 1) {
    if (t < off) red[t] += red[t + off];
    __syncthreads();
  }
  const float rn = 1.0f / sqrtf(red[0]);
  float* orow = s0 + (size_t)b * DDIM;
#pragma unroll
  for (int q = 0; q < 16; ++q) orow[t + (q << 8)] = v[q] * rn;
}

// --------------------------------------------------- CNOT-chain permutation
// perm[j] = p_0(p_1(...p_{n-1}(j))) ; p_i flips bit t when bit c is set.
static __device__ __forceinline__ int cnot_perm(int j) {
  int k = j;
  if (k & 1) k ^= (1 << (NW - 1));  // i = n-1: c = 0, t = n-1
#pragma unroll
  for (int c = 1; c < NW; ++c)      // i = n-2 .. 0: t = c-1
    if ((k >> c) & 1) k ^= (1 << (c - 1));
  return k;
}

__global__ __launch_bounds__(256) void perm_kernel(const float* __restrict__ ir,
                                                   const float* __restrict__ ii,
                                                   float* __restrict__ outr,
                                                   float* __restrict__ outi) {
  const int gid = blockIdx.x * 256 + threadIdx.x;
  const int b = gid >> NW;
  const int j = gid & (DDIM - 1);
  const int src = (b << NW) + cnot_perm(j);
  outr[gid] = ir[src];
  outi[gid] = ii[src];
}

// ---------------------------------------------------------------- |state|
__global__ __launch_bounds__(256) void abs_kernel(const float* __restrict__ r,
                                                  const float* __restrict__ i,
                                                  float* __restrict__ o) {
  const int gid = blockIdx.x * 256 + threadIdx.x;
  const float a = r[gid], b = i[gid];
  o[gid] = sqrtf(a * a + b * b);
}

// ----------------------------------------------------------- complex GEMM
// out[b][m] = sum_k U[m][k] * s[b][k]   (complex; CPLX=false => s imag == 0)
// WG: 16 output rows x full 256 batch cols, 8 waves, wave tile 16x32.
// Double-buffered LDS fed by GLOBAL_LOAD_ASYNC_TO_LDS (ASYNCcnt tracked).
template <bool CPLX>
__global__ __launch_bounds__(256) void gemm_kernel(
    const float* __restrict__ Ur, const float* __restrict__ Ui,
    const float* __restrict__ Sr, const float* __restrict__ Si,
    float* __restrict__ Or, float* __restrict__ Oi) {
  __shared__ float lds_u[2][2][16 * LDU];                      // [buf][r/i]
  __shared__ float lds_sr[2][NBAT * LDSS];
  __shared__ float lds_si[CPLX ? 2 : 1][CPLX ? NBAT * LDSS : 1];

  const int t = threadIdx.x;
  const int wave = t >> 5;
  const int lane = t & 31;
  const int hi = lane >> 4;        // half-wave select (K +2 / M +8)
  const int lo = lane & 15;        // row (A) / col (B,C)
  const int m0 = blockIdx.x << 4;  // this WG's 16 output rows
  const int colbase = wave << 5;   // this wave's 32 batch columns

  v8f accr[2] = {};
  v8f acci[2] = {};

  // Async-copy assignments (all lanes of every wave participate; the Ur/Ui
  // split is wave-uniform so EXEC stays all-ones for the async ops).
  const int mat = t >> 7;                 // 0: Ur (waves 0-3), 1: Ui (4-7)
  const int tt = t & 127;
  const int urow = tt >> 3;               // 0..15
  const int uj = (tt & 7) << 1;           // even float index 0..14
  const unsigned voffU = ((unsigned)(m0 + urow) * DDIM + (unsigned)uj) * 4u;
  const unsigned voffS = (unsigned)t * (DDIM * 4u);
  const uint64_t ubase = (uint64_t)(uintptr_t)(mat ? Ui : Ur);
  const uint64_t srbase = (uint64_t)(uintptr_t)Sr;
  const uint64_t sibase = CPLX ? (uint64_t)(uintptr_t)Si : 0;

  auto issue = [&](int k0, int nb) {
    const uint64_t koff = (uint64_t)k0 * 4u;
    const unsigned lu =
        (unsigned)(uintptr_t)&lds_u[nb][mat][urow * LDU + uj];
    ASYNC_CP64(lu, voffU, ubase + koff, 0);
    const unsigned lsr = (unsigned)(uintptr_t)&lds_sr[nb][t * LDSS];
    ASYNC_CP128(lsr, voffS, srbase + koff, 0);
    ASYNC_CP128(lsr, voffS, srbase + koff, 16);
    ASYNC_CP128(lsr, voffS, srbase + koff, 32);
    ASYNC_CP128(lsr, voffS, srbase + koff, 48);
    if constexpr (CPLX) {
      const unsigned lsi = (unsigned)(uintptr_t)&lds_si[nb][t * LDSS];
      ASYNC_CP128(lsi, voffS, sibase + koff, 0);
      ASYNC_CP128(lsi, voffS, sibase + koff, 16);
      ASYNC_CP128(lsi, voffS, sibase + koff, 32);
      ASYNC_CP128(lsi, voffS, sibase + koff, 48);
    }
  };

  issue(0, 0);
  for (int kt = 0; kt < DDIM / BK; ++kt) {
    // Outstanding asyncs == this chunk's only (next chunk not issued yet).
    WAIT_ASYNC0();
    __syncthreads();  // tile kt visible to all; buffer (kt+1)&1 free to fill
    if (kt + 1 < DDIM / BK) issue((kt + 1) * BK, (kt + 1) & 1);
    const int cb = kt & 1;

#pragma unroll
    for (int ks = 0; ks < 4; ++ks) {         // 4 WMMA k-steps per chunk
      const int kk = (ks << 2) + (hi << 1);  // A/B fragment K base per lane
      v2f ar = {lds_u[cb][0][lo * LDU + kk], lds_u[cb][0][lo * LDU + kk + 1]};
      v2f ai = {lds_u[cb][1][lo * LDU + kk], lds_u[cb][1][lo * LDU + kk + 1]};
      v2f nai = -ai;  // f32 WMMA NEG bits are C-only -> negate A_i in VALU
#pragma unroll
      for (int ct = 0; ct < 2; ++ct) {
        const int n = colbase + (ct << 4) + lo;
        v2f br = {lds_sr[cb][n * LDSS + kk], lds_sr[cb][n * LDSS + kk + 1]};
        accr[ct] = wmma_f32_4(ar, br, accr[ct]);  // Re += Ur*Sr
        acci[ct] = wmma_f32_4(ai, br, acci[ct]);  // Im += Ui*Sr
        if constexpr (CPLX) {
          v2f bi = {lds_si[cb][n * LDSS + kk], lds_si[cb][n * LDSS + kk + 1]};
          accr[ct] = wmma_f32_4(nai, bi, accr[ct]);  // Re -= Ui*Si
          acci[ct] = wmma_f32_4(ar, bi, acci[ct]);   // Im += Ur*Si
        }
      }
    }
  }

  // C/D layout: VGPR v of lane L holds M = v + 8*(L>>4), N = L&15
#pragma unroll
  for (int ct = 0; ct < 2; ++ct) {
    const int n = colbase + (ct << 4) + lo;
    const size_t base = (size_t)n * DDIM + m0 + (hi << 3);
    float4 r0 = make_float4(accr[ct][0], accr[ct][1], accr[ct][2], accr[ct][3]);
    float4 r1 = make_float4(accr[ct][4], accr[ct][5], accr[ct][6], accr[ct][7]);
    float4 i0 = make_float4(acci[ct][0], acci[ct][1], acci[ct][2], acci[ct][3]);
    float4 i1 = make_float4(acci[ct][4], acci[ct][5], acci[ct][6], acci[ct][7]);
    *(float4*)&Or[base] = r0;
    *(float4*)&Or[base + 4] = r1;
    *(float4*)&Oi[base] = i0;
    *(float4*)&Oi[base + 4] = i1;
  }
}

// ------------------------------------------------------------------ launch
extern "C" void kernel_launch(void* const* d_in, const int* in_sizes, int n_in,
                              void* d_out, int out_size, void* d_ws,
                              size_t ws_size, hipStream_t stream) {
  (void)in_sizes; (void)n_in; (void)out_size; (void)ws_size;
  const float* x = (const float*)d_in[0];
  const float* u0r = (const float*)d_in[1];
  const float* u0i = (const float*)d_in[2];
  const float* u1r = (const float*)d_in[3];
  const float* u1i = (const float*)d_in[4];
  const float* u2r = (const float*)d_in[5];
  const float* u2i = (const float*)d_in[6];
  const float* u3r = (const float*)d_in[7];
  const float* u3i = (const float*)d_in[8];
  float* out = (float*)d_out;

  const size_t BD = (size_t)NBAT * DDIM;  // 1M floats = 4 MB per buffer
  float* S0 = (float*)d_ws;               // needs 5 * 4 MB = 20 MB scratch
  float* Ar = S0 + BD;
  float* Ai = Ar + BD;
  float* Br = Ai + BD;
  float* Bi = Br + BD;

  dim3 blk(256);
  norm_kernel<<<NBAT, blk, 0, stream>>>(x, S0);
  gemm_kernel<false><<<DDIM / 16, blk, 0, stream>>>(u0r, u0i, S0, nullptr, Ar, Ai);
  perm_kernel<<<BD / 256, blk, 0, stream>>>(Ar, Ai, Br, Bi);
  gemm_kernel<true><<<DDIM / 16, blk, 0, stream>>>(u1r, u1i, Br, Bi, Ar, Ai);
  gemm_kernel<true><<<DDIM / 16, blk, 0, stream>>>(u2r, u2i, Ar, Ai, Br, Bi);
  perm_kernel<<<BD / 256, blk, 0, stream>>>(Br, Bi, Ar, Ai);
  gemm_kernel<true><<<DDIM / 16, blk, 0, stream>>>(u3r, u3i, Ar, Ai, Br, Bi);
  abs_kernel<<<BD / 256, blk, 0, stream>>>(Br, Bi, out);
}